// AdaConv2D_8821862826746
// MI455X (gfx1250) — compile-verified
//
#include <hip/hip_runtime.h>
#include <hip/hip_bf16.h>

// Problem constants (from reference setup_inputs)
#define BB   16
#define CC   256
#define OO   256
#define HH   128
#define WW   128
#define GG   32
#define CPG  8     // channels per group
#define OPG  8     // outputs per group
#define KTOT 72    // 8 in-ch * 3 * 3
#define NPIX (HH*WW)

typedef __attribute__((ext_vector_type(2))) float v2f;
typedef __attribute__((ext_vector_type(8))) float v8f;

// ---------------------------------------------------------------------------
// Kernel 1: per-(b,c) instance-norm statistics.
// mean[bc], istd[bc] = 1/(sqrt(unbiased var)+1e-7)
// ---------------------------------------------------------------------------
__global__ __launch_bounds__(256) void in_stats_kernel(
    const float* __restrict__ x, float* __restrict__ mean, float* __restrict__ istd)
{
    const int bc = blockIdx.x;                 // 0 .. B*C-1
    const float* p = x + (size_t)bc * NPIX;
    float s = 0.f, s2 = 0.f;
    for (int i = threadIdx.x; i < NPIX; i += 256) {
        float v = p[i];
        s  += v;
        s2 += v * v;
    }
    __shared__ float rs[256];
    __shared__ float rq[256];
    rs[threadIdx.x] = s;
    rq[threadIdx.x] = s2;
    __syncthreads();
    for (int off = 128; off > 0; off >>= 1) {
        if ((int)threadIdx.x < off) {
            rs[threadIdx.x] += rs[threadIdx.x + off];
            rq[threadIdx.x] += rq[threadIdx.x + off];
        }
        __syncthreads();
    }
    if (threadIdx.x == 0) {
        const float N  = (float)NPIX;
        float mu  = rs[0] / N;
        float var = (rq[0] - N * mu * mu) / (N - 1.0f);   // ddof=1 (torch.std)
        var = fmaxf(var, 0.f);
        float sd = sqrtf(var) + 1e-7f;                    // eps on std, not var
        mean[bc] = mu;
        istd[bc] = 1.0f / sd;
    }
}

// ---------------------------------------------------------------------------
// Kernel 2: fold pointwise 1x1 group conv into the depthwise-group 3x3 kernel.
// Wc[b,g][o][k] = sum_i pw[b][g*8+o][i] * dw[b][g*8+i][k],  k in [0,72)
// ---------------------------------------------------------------------------
__global__ __launch_bounds__(64) void wcomb_kernel(
    const float* __restrict__ dw, const float* __restrict__ pw, float* __restrict__ wc)
{
    const int bg = blockIdx.x;                 // 0 .. B*G-1
    const int b = bg >> 5;
    const int g = bg & 31;
    const float* dwp = dw + (size_t)(b * OO + g * OPG) * KTOT;  // [i][k], i stride KTOT
    const float* pwp = pw + (size_t)(b * OO + g * OPG) * CPG;   // [o][i], o stride 8
    float* wcp = wc + (size_t)bg * (OPG * KTOT);
    for (int e = threadIdx.x; e < OPG * KTOT; e += 64) {
        const int o = e / KTOT;
        const int k = e - o * KTOT;
        float acc = 0.f;
#pragma unroll
        for (int i = 0; i < 8; ++i)
            acc += pwp[o * 8 + i] * dwp[i * KTOT + k];
        wcp[e] = acc;
    }
}

// ---------------------------------------------------------------------------
// Kernel 3: fused norm + grouped 3x3 conv via V_WMMA_F32_16X16X4_F32.
// Block = (b, g, 8-row strip). 256 threads = 8 wave32.
// Per wave: one output row, 8 column-tiles of 16 pixels.
// WMMA tile: A = Wc (16x4 slice, rows 8..15 zero), B = im2col (4x16 pixels),
//            C/D = out channels x pixels (f32).
// Lane layout (ISA 7.12.2, 32-bit): for A and B, (vgpr v, lane) -> K = 2*(lane>>4)+v;
// A row M = lane&15, B col N = lane&15. C/D: VGPR r -> M = r + 8*(lane>>4), N = lane&15.
// ---------------------------------------------------------------------------
#define LDSW 132   // 1 zero + 128 + 1 zero, padded to 132 for bank spread

__global__ __launch_bounds__(256) void adaconv_wmma_kernel(
    const float* __restrict__ x,
    const float* __restrict__ mean, const float* __restrict__ istd,
    const float* __restrict__ wc,   const float* __restrict__ bias,
    float* __restrict__ out)
{
    __shared__ float sx[CPG][10][LDSW];   // normalized input halo tile
    __shared__ float sw[OPG][KTOT];       // combined weights
    __shared__ float sb[OPG];             // biases

    const int blk = blockIdx.x;           // ((b*32 + g)*16 + hb)
    const int hb  = blk & 15;
    const int g   = (blk >> 4) & 31;
    const int b   = blk >> 9;
    const int h0  = hb * 8;
    const int tid = threadIdx.x;

    // --- stage weights / bias ---
    {
        const float* wcp = wc + (size_t)(b * GG + g) * (OPG * KTOT);
        for (int i = tid; i < OPG * KTOT; i += 256)
            sw[i / KTOT][i % KTOT] = wcp[i];
        if (tid < OPG)
            sb[tid] = bias[b * OO + g * OPG + tid];
        if (tid < CPG * 10) {             // zero the W-halo columns
            int ch = tid / 10, r = tid % 10;
            sx[ch][r][0]   = 0.f;
            sx[ch][r][129] = 0.f;
        }
    }

    // --- stage normalized input: 8 ch x 10 rows (h0-1 .. h0+8) x 128 cols ---
    for (int i = tid; i < CPG * 10 * WW; i += 256) {
        const int col = i & (WW - 1);
        const int r   = (i >> 7) % 10;
        const int ch  = i / (WW * 10);
        const int gh  = h0 - 1 + r;
        const int c   = g * CPG + ch;
        float v = 0.f;
        if (gh >= 0 && gh < HH) {
            const float xv = x[(((size_t)b * CC + c) * HH + gh) * WW + col];
            v = (xv - mean[b * CC + c]) * istd[b * CC + c];
        }
        sx[ch][r][col + 1] = v;
    }
    __syncthreads();

    const int lane = tid & 31;
    const int wave = tid >> 5;
    const int half = lane >> 4;   // selects K pair within fragment
    const int n    = lane & 15;   // B column / A row / C column

    // --- A fragments: constant across all tiles of this block ---
    v2f afrag[18];
#pragma unroll
    for (int s = 0; s < 18; ++s) {
#pragma unroll
        for (int v = 0; v < 2; ++v) {
            const int k = 4 * s + 2 * half + v;        // 0..71
            afrag[s][v] = (n < OPG) ? sw[n][k] : 0.f;  // rows 8..15 zero-pad
        }
    }

    const int r = wave;                    // local output row 0..7
    for (int ct = 0; ct < WW / 16; ++ct) { // 8 column tiles
        const int cb = ct * 16;
        v8f acc = {};
#pragma unroll
        for (int s = 0; s < 18; ++s) {
            v2f bfrag;
#pragma unroll
            for (int v = 0; v < 2; ++v) {
                const int k   = 4 * s + 2 * half + v;
                const int ci  = k / 9;
                const int rem = k - ci * 9;
                const int kh  = rem / 3;
                const int kw  = rem - kh * 3;
                // col index: 1 + (cb + n + kw - 1) = cb + n + kw  (halo pad)
                bfrag[v] = sx[ci][r + kh][cb + n + kw];
            }
            acc = __builtin_amdgcn_wmma_f32_16x16x4_f32(
                false, afrag[s], false, bfrag, (short)0, acc, false, false);
        }
        // D layout: lanes 0..15 hold M=0..7 (valid), lanes 16..31 hold M=8..15 (pad)
        if (lane < 16) {
            const int col = cb + n;
            const int hrow = h0 + r;
#pragma unroll
            for (int m = 0; m < OPG; ++m) {
                out[(((size_t)b * OO + g * OPG + m) * HH + hrow) * WW + col]
                    = acc[m] + sb[m];
            }
        }
    }
}

// ---------------------------------------------------------------------------
// Launch
// ---------------------------------------------------------------------------
extern "C" void kernel_launch(void* const* d_in, const int* in_sizes, int n_in,
                              void* d_out, int out_size, void* d_ws, size_t ws_size,
                              hipStream_t stream)
{
    const float* x    = (const float*)d_in[0];   // [B,C,H,W]
    const float* dw   = (const float*)d_in[1];   // [B,O,8,3,3]
    const float* pw   = (const float*)d_in[2];   // [B,O,8,1,1]
    const float* bias = (const float*)d_in[3];   // [B,O]
    float* out = (float*)d_out;

    // workspace layout (floats): mean[4096] | istd[4096] | Wc[512*576]
    float* wsf  = (float*)d_ws;
    float* mean = wsf;
    float* istd = wsf + BB * CC;
    float* wcmb = wsf + 2 * BB * CC;

    in_stats_kernel<<<BB * CC, 256, 0, stream>>>(x, mean, istd);
    wcomb_kernel<<<BB * GG, 64, 0, stream>>>(dw, pw, wcmb);
    adaconv_wmma_kernel<<<BB * GG * (HH / 8), 256, 0, stream>>>(
        x, mean, istd, wcmb, bias, out);
}